// chamferDist_43800076484722
// MI455X (gfx1250) — compile-verified
//
#include <hip/hip_runtime.h>
#include <hip/hip_bf16.h>

typedef __attribute__((ext_vector_type(2))) float v2f;
typedef __attribute__((ext_vector_type(8))) float v8f;

#define WAVES_PER_BLOCK 8

// Computes out[b, i] = min_j ( |P_i|^2 + |Q_j|^2 - 2 <P_i, Q_j> )
// One wave handles a 16-row tile of P and streams all of Q in 16-col tiles.
// Each tile is a single V_WMMA_F32_16X16X4_F32 with:
//   A = -2 * P-tile (16x4, K=3 padded with 0), loaded once
//   B =      Q-tile (4x16)
//   C = |Q_j|^2 broadcast down columns
// so D = |Q_j|^2 - 2<P_i,Q_j> directly; running row-min is 8 v_min per tile.
__global__ __launch_bounds__(256)
void chamfer_min_wmma(const float* __restrict__ P, int nP,
                      const float* __restrict__ Q, int nQ,
                      float* __restrict__ out) {
  const int lane = threadIdx.x & 31;
  const int wave = threadIdx.x >> 5;
  const int b    = blockIdx.y;
  const int half = lane >> 4;   // 0: lanes 0-15, 1: lanes 16-31
  const int l16  = lane & 15;

  const int ibase = (blockIdx.x * WAVES_PER_BLOCK + wave) * 16;
  if (ibase >= nP) return;      // wave-uniform; EXEC stays all-1s for WMMA

  const float* __restrict__ Pb = P + (size_t)b * nP * 3;
  const float* __restrict__ Qb = Q + (size_t)b * nQ * 3;

  // ---- Load A tile (16 rows of P). ISA layout for 32-bit A 16x4:
  //   lanes 0-15:  VGPR0 = K=0, VGPR1 = K=1
  //   lanes 16-31: VGPR0 = K=2, VGPR1 = K=3
  const int prow = ibase + l16;
  const float px = Pb[prow * 3 + 0];
  const float py = Pb[prow * 3 + 1];
  const float pz = Pb[prow * 3 + 2];
  const float sq1 = fmaf(px, px, fmaf(py, py, pz * pz));  // |P_i|^2, lane l16 & l16+16
  v2f a;
  a[0] = half ? (-2.0f * pz) : (-2.0f * px);
  a[1] = half ? 0.0f         : (-2.0f * py);

  v8f rm;
  #pragma unroll
  for (int v = 0; v < 8; ++v) rm[v] = 3.0e38f;

  for (int j = 0; j < nQ; j += 16) {
    // B tile (4x16): N across lanes (col = l16 in both halves), K split by half
    const int qrow = j + l16;
    const float qx = Qb[qrow * 3 + 0];
    const float qy = Qb[qrow * 3 + 1];
    const float qz = Qb[qrow * 3 + 2];
    const float sq2 = fmaf(qx, qx, fmaf(qy, qy, qz * qz)); // |Q_j|^2 for col l16
    v2f bb;
    bb[0] = half ? qz   : qx;
    bb[1] = half ? 0.0f : qy;

    v8f c;
    #pragma unroll
    for (int v = 0; v < 8; ++v) c[v] = sq2;  // C[v][lane] = |Q_{col}|^2

    // D = A*B + C = |Q_j|^2 - 2<P_i,Q_j>
    v8f d = __builtin_amdgcn_wmma_f32_16x16x4_f32(
        false, a, false, bb, (short)0, c, false, false);

    #pragma unroll
    for (int v = 0; v < 8; ++v) rm[v] = fminf(rm[v], d[v]);
  }

  // ---- Row-min across the 16 lanes of each half (rows stay within a half:
  // lanes 0-15 hold M = v, lanes 16-31 hold M = v+8)
  #pragma unroll
  for (int off = 1; off < 16; off <<= 1) {
    #pragma unroll
    for (int v = 0; v < 8; ++v) {
      rm[v] = fminf(rm[v], __shfl_xor(rm[v], off, 32));
    }
  }

  // ---- Add |P_i|^2 and write. Lane 0 writes rows 0..7, lane 16 rows 8..15.
  float* __restrict__ outb = out + (size_t)b * nP;
  #pragma unroll
  for (int v = 0; v < 8; ++v) {
    const int srcRow = v + (half << 3);             // row within this tile
    const float s1 = __shfl(sq1, srcRow, 32);       // lane srcRow holds |P_srcRow|^2
    const float res = rm[v] + s1;
    if (l16 == 0) outb[ibase + srcRow] = res;
  }
}

extern "C" void kernel_launch(void* const* d_in, const int* in_sizes, int n_in,
                              void* d_out, int out_size, void* d_ws, size_t ws_size,
                              hipStream_t stream) {
  const float* xyz1 = (const float*)d_in[0];
  const float* xyz2 = (const float*)d_in[1];
  float* out = (float*)d_out;

  const int B = 4, D = 3;
  const int N = in_sizes[0] / (B * D);
  const int M = in_sizes[1] / (B * D);

  dim3 blk(32 * WAVES_PER_BLOCK);

  // Pass 1: dist1[b, i] = min_j d2  (scan xyz2 for every xyz1 row tile)
  dim3 g1((N + 16 * WAVES_PER_BLOCK - 1) / (16 * WAVES_PER_BLOCK), B);
  chamfer_min_wmma<<<g1, blk, 0, stream>>>(xyz1, N, xyz2, M, out);

  // Pass 2: dist2[b, j] = min_i d2  (roles swapped; structure is symmetric)
  dim3 g2((M + 16 * WAVES_PER_BLOCK - 1) / (16 * WAVES_PER_BLOCK), B);
  chamfer_min_wmma<<<g2, blk, 0, stream>>>(xyz2, M, xyz1, N, out + (size_t)B * N);
}